// Mismatch_loss_19018115187338
// MI455X (gfx1250) — compile-verified
//
#include <hip/hip_runtime.h>
#include <hip/hip_bf16.h>

typedef __attribute__((ext_vector_type(2))) float v2f;
typedef __attribute__((ext_vector_type(8))) float v8f;

#define BDIM 256
#define NPAIR 16
#define NBIN 8192
#define V_PER_PAIR (1u << 21)
#define VEC4_PER_PAIR (V_PER_PAIR / 4)          // 524288 float4 per pair
#define F4_PER_THREAD 8
#define BLOCKS_X (VEC4_PER_PAIR / (BDIM * F4_PER_THREAD))  // 256
#define K_SEL 209715u                            // int(128^3 * 10 / 100)

// ---------------- workspace zeroing ----------------
__global__ void zero_ws(unsigned* __restrict__ w, int n) {
    int i = blockIdx.x * blockDim.x + threadIdx.x;
    if (i < n) w[i] = 0u;
}

__device__ __forceinline__ float ce_of(float p, float t) {
    // ce = -(t * log(p)) >= 0 ; clamp away any -0.0 corner
    return fmaxf(-(t * logf(p)), 0.0f);
}

// wave32 sum via V_WMMA_F32_16X16X4_F32:
// A[m][0]=partial(lane m), A[m][2]=partial(lane m+16), B=ones ->
// D[m][n] = partial[m]+partial[m+16]; lane-local sum of 8 D regs gives
// half-wave sums; one shfl_xor(16) completes the 32-lane sum.
// Must be called with all lanes active (EXEC all-ones).
__device__ __forceinline__ float wave_sum_wmma(float ls) {
    v2f a; a[0] = ls;   a[1] = 0.0f;
    v2f b; b[0] = 1.0f; b[1] = 1.0f;
    v8f c = {};
    v8f d = __builtin_amdgcn_wmma_f32_16x16x4_f32(false, a, false, b, (short)0, c, false, false);
    float s = d[0] + d[1] + d[2] + d[3] + d[4] + d[5] + d[6] + d[7];
    s += __shfl_xor(s, 16, 32);
    return s;
}

// ---------------- pass A: 13-bit histogram + activity maxima (+ optional ce-bits cache) ----------------
__global__ void histA(const float4* __restrict__ net, const float4* __restrict__ tgt,
                      const float4* __restrict__ mpos, unsigned* __restrict__ hist1,
                      unsigned* __restrict__ maxT, unsigned* __restrict__ maxP,
                      uint4* __restrict__ cebits) {
    __shared__ unsigned h[NBIN];
    __shared__ unsigned smT, smP;
    int t = threadIdx.x;
    for (int i = t; i < NBIN; i += BDIM) h[i] = 0u;
    if (t == 0) { smT = 0u; smP = 0u; }
    __syncthreads();

    int p = blockIdx.y;
    unsigned base = (unsigned)p * VEC4_PER_PAIR + blockIdx.x * (BDIM * F4_PER_THREAD) + t;
    unsigned lmT = 0u, lmP = 0u;
    for (int i = 0; i < F4_PER_THREAD; ++i) {
        unsigned idx = base + i * BDIM;
        float4 n = net[idx];
        float4 g = tgt[idx];
        float4 m = mpos[idx];
        uint4 bb;
        bb.x = __float_as_uint(ce_of(n.x, g.x));
        bb.y = __float_as_uint(ce_of(n.y, g.y));
        bb.z = __float_as_uint(ce_of(n.z, g.z));
        bb.w = __float_as_uint(ce_of(n.w, g.w));
        atomicAdd(&h[bb.x >> 19], 1u);
        atomicAdd(&h[bb.y >> 19], 1u);
        atomicAdd(&h[bb.z >> 19], 1u);
        atomicAdd(&h[bb.w >> 19], 1u);
        lmT = max(lmT, max(max(__float_as_uint(g.x), __float_as_uint(g.y)),
                           max(__float_as_uint(g.z), __float_as_uint(g.w))));
        lmP = max(lmP, max(max(__float_as_uint(m.x), __float_as_uint(m.y)),
                           max(__float_as_uint(m.z), __float_as_uint(m.w))));
        if (cebits) cebits[idx] = bb;  // uniform branch
    }
    atomicMax(&smT, lmT);
    atomicMax(&smP, lmP);
    __syncthreads();

    unsigned* gh = hist1 + p * NBIN;
    for (int i = t; i < NBIN; i += BDIM) {
        unsigned c = h[i];
        if (c) atomicAdd(&gh[i], c);
    }
    if (t == 0) {
        atomicMax(&maxT[p], smT);
        atomicMax(&maxP[p], smP);
    }
}

// ---------------- top-down rank scan over an 8192-bin histogram ----------------
__global__ void scan_hist(const unsigned* __restrict__ hist, const unsigned* __restrict__ rank_in,
                          unsigned rank_const, unsigned* __restrict__ selBin,
                          unsigned* __restrict__ rank_out) {
    __shared__ unsigned cs[BDIM];
    int p = blockIdx.x;
    int t = threadIdx.x;
    const unsigned* hp = hist + p * NBIN;
    unsigned k = rank_in ? rank_in[p] : rank_const;

    int base = t * 32;  // bins in DESCENDING order: bin = NBIN-1-(base+j)
    unsigned s = 0u;
    for (int j = 0; j < 32; ++j) s += hp[NBIN - 1 - (base + j)];
    cs[t] = s;
    __syncthreads();
    for (int off = 1; off < BDIM; off <<= 1) {
        unsigned v = (t >= off) ? cs[t - off] : 0u;
        __syncthreads();
        cs[t] += v;
        __syncthreads();
    }
    unsigned cumBefore = cs[t] - s;  // counts in bins strictly above my chunk
    if (cumBefore < k && k <= cumBefore + s) {
        unsigned c = cumBefore;
        for (int j = 0; j < 32; ++j) {
            int b = NBIN - 1 - (base + j);
            unsigned hv = hp[b];
            if (c + hv >= k) {
                selBin[p] = (unsigned)b;
                rank_out[p] = k - c;  // in [1, hp[b]]
                break;
            }
            c += hv;
        }
    }
}

// ---------------- pass B (two variants): refine next 13 bits within selected bin ----------------
__device__ __forceinline__ void histB_accum(unsigned bits, unsigned sel, unsigned* h) {
    if ((bits >> 19) == sel) atomicAdd(&h[(bits >> 6) & (NBIN - 1)], 1u);
}

__global__ void histB_bits(const uint4* __restrict__ cebits, const unsigned* __restrict__ b1,
                           unsigned* __restrict__ hist2) {
    __shared__ unsigned h[NBIN];
    int t = threadIdx.x;
    for (int i = t; i < NBIN; i += BDIM) h[i] = 0u;
    __syncthreads();
    int p = blockIdx.y;
    unsigned sel = b1[p];
    unsigned base = (unsigned)p * VEC4_PER_PAIR + blockIdx.x * (BDIM * F4_PER_THREAD) + t;
    for (int i = 0; i < F4_PER_THREAD; ++i) {
        uint4 bb = cebits[base + i * BDIM];
        histB_accum(bb.x, sel, h);
        histB_accum(bb.y, sel, h);
        histB_accum(bb.z, sel, h);
        histB_accum(bb.w, sel, h);
    }
    __syncthreads();
    unsigned* gh = hist2 + p * NBIN;
    for (int i = t; i < NBIN; i += BDIM) {
        unsigned c = h[i];
        if (c) atomicAdd(&gh[i], c);
    }
}

__global__ void histB_recompute(const float4* __restrict__ net, const float4* __restrict__ tgt,
                                const unsigned* __restrict__ b1, unsigned* __restrict__ hist2) {
    __shared__ unsigned h[NBIN];
    int t = threadIdx.x;
    for (int i = t; i < NBIN; i += BDIM) h[i] = 0u;
    __syncthreads();
    int p = blockIdx.y;
    unsigned sel = b1[p];
    unsigned base = (unsigned)p * VEC4_PER_PAIR + blockIdx.x * (BDIM * F4_PER_THREAD) + t;
    for (int i = 0; i < F4_PER_THREAD; ++i) {
        unsigned idx = base + i * BDIM;
        float4 n = net[idx];
        float4 g = tgt[idx];
        histB_accum(__float_as_uint(ce_of(n.x, g.x)), sel, h);
        histB_accum(__float_as_uint(ce_of(n.y, g.y)), sel, h);
        histB_accum(__float_as_uint(ce_of(n.z, g.z)), sel, h);
        histB_accum(__float_as_uint(ce_of(n.w, g.w)), sel, h);
    }
    __syncthreads();
    unsigned* gh = hist2 + p * NBIN;
    for (int i = t; i < NBIN; i += BDIM) {
        unsigned c = h[i];
        if (c) atomicAdd(&gh[i], c);
    }
}

// ---------------- pass C (two variants): sum above 26-bit prefix + count ties ----------------
__device__ __forceinline__ void sumC_accum(unsigned bits, unsigned P26, float& ls, unsigned* h64) {
    unsigned pr = bits >> 6;
    if (pr > P26) ls += __uint_as_float(bits);
    else if (pr == P26) atomicAdd(&h64[bits & 63u], 1u);
}

__device__ __forceinline__ void sumC_tail(float ls, int p, int t, unsigned* h64,
                                          float* wsum, unsigned* cnt64, double* gSum) {
    float s = wave_sum_wmma(ls);  // all lanes active here
    int lane = t & 31, w = t >> 5;
    if (lane == 0) wsum[w] = s;
    __syncthreads();
    if (t == 0) {
        float bs = 0.0f;
        for (int i = 0; i < BDIM / 32; ++i) bs += wsum[i];
        atomicAdd(&gSum[p], (double)bs);
    }
    if (t < 64) {
        unsigned cc = h64[t];
        if (cc) atomicAdd(&cnt64[p * 64 + t], cc);
    }
}

__global__ void sumC_bits(const uint4* __restrict__ cebits, const unsigned* __restrict__ b1,
                          const unsigned* __restrict__ b2, unsigned* __restrict__ cnt64,
                          double* __restrict__ gSum) {
    __shared__ unsigned h64[64];
    __shared__ float wsum[BDIM / 32];
    int t = threadIdx.x;
    if (t < 64) h64[t] = 0u;
    __syncthreads();
    int p = blockIdx.y;
    unsigned P26 = (b1[p] << 13) | b2[p];
    unsigned base = (unsigned)p * VEC4_PER_PAIR + blockIdx.x * (BDIM * F4_PER_THREAD) + t;
    float ls = 0.0f;
    for (int i = 0; i < F4_PER_THREAD; ++i) {
        uint4 bb = cebits[base + i * BDIM];
        sumC_accum(bb.x, P26, ls, h64);
        sumC_accum(bb.y, P26, ls, h64);
        sumC_accum(bb.z, P26, ls, h64);
        sumC_accum(bb.w, P26, ls, h64);
    }
    sumC_tail(ls, p, t, h64, wsum, cnt64, gSum);
}

__global__ void sumC_recompute(const float4* __restrict__ net, const float4* __restrict__ tgt,
                               const unsigned* __restrict__ b1, const unsigned* __restrict__ b2,
                               unsigned* __restrict__ cnt64, double* __restrict__ gSum) {
    __shared__ unsigned h64[64];
    __shared__ float wsum[BDIM / 32];
    int t = threadIdx.x;
    if (t < 64) h64[t] = 0u;
    __syncthreads();
    int p = blockIdx.y;
    unsigned P26 = (b1[p] << 13) | b2[p];
    unsigned base = (unsigned)p * VEC4_PER_PAIR + blockIdx.x * (BDIM * F4_PER_THREAD) + t;
    float ls = 0.0f;
    for (int i = 0; i < F4_PER_THREAD; ++i) {
        unsigned idx = base + i * BDIM;
        float4 n = net[idx];
        float4 g = tgt[idx];
        sumC_accum(__float_as_uint(ce_of(n.x, g.x)), P26, ls, h64);
        sumC_accum(__float_as_uint(ce_of(n.y, g.y)), P26, ls, h64);
        sumC_accum(__float_as_uint(ce_of(n.z, g.z)), P26, ls, h64);
        sumC_accum(__float_as_uint(ce_of(n.w, g.w)), P26, ls, h64);
    }
    sumC_tail(ls, p, t, h64, wsum, cnt64, gSum);
}

// ---------------- finalize: exact tie handling + per-image aggregation ----------------
__global__ void finalize(const double* __restrict__ gSum, const unsigned* __restrict__ cnt64,
                         const unsigned* __restrict__ b1, const unsigned* __restrict__ b2,
                         const unsigned* __restrict__ r2, const unsigned* __restrict__ maxT,
                         const unsigned* __restrict__ maxP, float* __restrict__ out) {
    __shared__ float losses[NPAIR];
    int t = threadIdx.x;
    if (t < NPAIR) {
        unsigned P26 = (b1[t] << 13) | b2[t];
        unsigned rem = r2[t];
        double ssel = 0.0;
        for (int bb = 63; bb >= 0 && rem > 0u; --bb) {
            unsigned c = cnt64[t * 64 + bb];
            unsigned m = c < rem ? c : rem;
            if (m) {
                // all values in this bin share the same full 32-bit pattern
                float v = __uint_as_float((P26 << 6) | (unsigned)bb);
                ssel += (double)v * (double)m;
                rem -= m;
            }
        }
        double loss = (gSum[t] + ssel) / (double)K_SEL;
        bool active = !((maxT[t] == 0u) && (maxP[t] == 0u));
        losses[t] = active ? (float)loss : 0.0f;
    }
    __syncthreads();
    if (t == 0) {
        float tot = 0.0f;
        for (int img = 0; img < 4; ++img) {
            float s = 0.0f;
            int c = 0;
            for (int k = 0; k < 4; ++k) {
                float v = losses[img * 4 + k];
                s += v;
                if (v != 0.0f) c++;
            }
            tot += s / (float)c;
        }
        out[0] = tot / 4.0f;
    }
}

extern "C" void kernel_launch(void* const* d_in, const int* in_sizes, int n_in,
                              void* d_out, int out_size, void* d_ws, size_t ws_size,
                              hipStream_t stream) {
    const float4* net = (const float4*)d_in[0];
    const float4* tgt = (const float4*)d_in[1];
    const float4* mpos = (const float4*)d_in[2];
    float* out = (float*)d_out;

    // small workspace region (~1.03 MB)
    unsigned* hist1 = (unsigned*)d_ws;            // 16*8192
    unsigned* hist2 = hist1 + NPAIR * NBIN;       // 16*8192
    unsigned* cnt64 = hist2 + NPAIR * NBIN;       // 16*64
    unsigned* maxT = cnt64 + NPAIR * 64;          // 16
    unsigned* maxP = maxT + NPAIR;                // 16
    unsigned* b1 = maxP + NPAIR;                  // 16
    unsigned* r1 = b1 + NPAIR;                    // 16
    unsigned* b2 = r1 + NPAIR;                    // 16
    unsigned* r2 = b2 + NPAIR;                    // 16
    double* gSum = (double*)(r2 + NPAIR);         // 16 doubles

    const int zwords = NPAIR * NBIN * 2 + NPAIR * 64 + NPAIR * 6 + NPAIR * 2;
    const size_t small_bytes = (size_t)zwords * 4u;          // 1,053,184 B (256-aligned)
    const size_t cebits_bytes = (size_t)NPAIR * V_PER_PAIR * 4u;  // 134 MB
    bool use_bits = (ws_size >= small_bytes + cebits_bytes);
    uint4* cebits = use_bits ? (uint4*)((char*)d_ws + small_bytes) : nullptr;

    zero_ws<<<(zwords + BDIM - 1) / BDIM, BDIM, 0, stream>>>((unsigned*)d_ws, zwords);

    dim3 grid(BLOCKS_X, NPAIR);
    histA<<<grid, BDIM, 0, stream>>>(net, tgt, mpos, hist1, maxT, maxP, cebits);
    scan_hist<<<NPAIR, BDIM, 0, stream>>>(hist1, nullptr, K_SEL, b1, r1);
    if (use_bits) {
        histB_bits<<<grid, BDIM, 0, stream>>>(cebits, b1, hist2);
    } else {
        histB_recompute<<<grid, BDIM, 0, stream>>>(net, tgt, b1, hist2);
    }
    scan_hist<<<NPAIR, BDIM, 0, stream>>>(hist2, r1, 0u, b2, r2);
    if (use_bits) {
        sumC_bits<<<grid, BDIM, 0, stream>>>(cebits, b1, b2, cnt64, gSum);
    } else {
        sumC_recompute<<<grid, BDIM, 0, stream>>>(net, tgt, b1, b2, cnt64, gSum);
    }
    finalize<<<1, 32, 0, stream>>>(gSum, cnt64, b1, b2, r2, maxT, maxP, out);
}